// RewardModel_56246891708911
// MI455X (gfx1250) — compile-verified
//
#include <hip/hip_runtime.h>
#include <math.h>

#define BPAIRS 32
#define SEQ    2048
#define HID    1024

typedef __attribute__((ext_vector_type(2))) float v2f;
typedef __attribute__((ext_vector_type(8))) float v8f;

// ---------------------------------------------------------------------------
// Phase 1: rewards[row] = dot(hidden[row, :], w) for row in [0, 2B*S)
// One wave (32 lanes) per 16-row tile, mapped onto V_WMMA_F32_16X16X4_F32.
// A tile (16x4 f32): lane L<16 holds rows M=L, K=0..1; lane 16+L holds K=2..3.
// B tile (4x16 f32): VGPR v = K {v | v+2} per half-wave; we broadcast w across
// all 16 N columns, so B's per-lane k-offset is identical to A's.
// ---------------------------------------------------------------------------
__global__ __launch_bounds__(256) void gemv_rewards_wmma(
    const float* __restrict__ hidden, const float* __restrict__ w,
    float* __restrict__ rewards) {
  const int lane = threadIdx.x & 31;
  const int wave = threadIdx.x >> 5;
  const int tile = blockIdx.x * 8 + wave;        // 16 rows per tile
  const int r    = lane & 15;                    // row within tile (A matrix M)
  const int kh   = (lane >> 4) << 1;             // k sub-offset per half-wave

  const float* rowp = hidden + (size_t)(tile * 16 + r) * HID + kh;
  const float* wp   = w + kh;

  v8f acc0 = {};
  v8f acc1 = {};
  #pragma unroll 4
  for (int k = 0; k < HID; k += 8) {
    v2f a0 = *(const v2f*)(rowp + k);
    v2f b0 = *(const v2f*)(wp + k);
    v2f a1 = *(const v2f*)(rowp + k + 4);
    v2f b1 = *(const v2f*)(wp + k + 4);
    acc0 = __builtin_amdgcn_wmma_f32_16x16x4_f32(false, a0, false, b0,
                                                 (short)0, acc0, false, false);
    acc1 = __builtin_amdgcn_wmma_f32_16x16x4_f32(false, a1, false, b1,
                                                 (short)0, acc1, false, false);
  }
  v8f acc = acc0 + acc1;

  // C/D layout: VGPR v, lanes 0-15 -> M=v (N=lane); lanes 16-31 -> M=v+8.
  // Column N==0 (lane 0) and N==0 of upper half (lane 16) carry the results.
  if ((lane & 15) == 0) {
    const int mbase = tile * 16 + (lane >> 4) * 8;
    #pragma unroll
    for (int v = 0; v < 8; ++v) rewards[mbase + v] = acc[v];
  }
}

// ---------------------------------------------------------------------------
// Phase 2: per-pair index logic + masked reductions. One block per pair.
// ---------------------------------------------------------------------------
__device__ __forceinline__ int block_min_i(int v, int tid, int* sh) {
  sh[tid] = v; __syncthreads();
  for (int o = 128; o > 0; o >>= 1) {
    if (tid < o) sh[tid] = min(sh[tid], sh[tid + o]);
    __syncthreads();
  }
  int r = sh[0]; __syncthreads();
  return r;
}

__device__ __forceinline__ float block_sum_f(float v, int tid, float* sh) {
  sh[tid] = v; __syncthreads();
  for (int o = 128; o > 0; o >>= 1) {
    if (tid < o) sh[tid] += sh[tid + o];
    __syncthreads();
  }
  float r = sh[0]; __syncthreads();
  return r;
}

__global__ __launch_bounds__(256) void pair_reduce(
    const int* __restrict__ ids, const float* __restrict__ rewards,
    float* __restrict__ out, float* __restrict__ partials) {
  const int b   = blockIdx.x;
  const int tid = threadIdx.x;
  const int*   ic = ids     + (size_t)b * SEQ;
  const int*   ir = ids     + (size_t)(BPAIRS + b) * SEQ;
  const float* rc = rewards + (size_t)b * SEQ;
  const float* rr = rewards + (size_t)(BPAIRS + b) * SEQ;

  __shared__ int   sh_i[256];
  __shared__ float sh_f[256];

  int tpc = SEQ, tpr = SEQ, tdf = SEQ;
  for (int s = tid; s < SEQ; s += 256) {
    int a = ic[s], c = ir[s];
    if (a == 0 && s < tpc) tpc = s;
    if (c == 0 && s < tpr) tpr = s;
    if (a != c && s < tdf) tdf = s;
  }
  const int c_ind     = block_min_i(tpc, tid, sh_i);  // first PAD in chosen (or SEQ)
  const int r_ind_pad = block_min_i(tpr, tid, sh_i);  // first PAD in rejected (or SEQ)
  const int mindiff   = block_min_i(tdf, tid, sh_i);  // first divergence (or SEQ)

  const bool has_div = (mindiff < SEQ);
  const int  div_ind = has_div ? mindiff : (SEQ - 1);
  const int  r_ind   = has_div ? r_ind_pad : c_ind;
  const int  end_ind = has_div ? max(c_ind, r_ind_pad) : SEQ;

  float ls = 0.f, sc = 0.f, sr = 0.f;
  for (int s = div_ind + tid; s < end_ind; s += 256) {
    float zc = rc[s], zr = rr[s];
    float z = zc - zr;
    // stable log-sigmoid: min(z,0) - log1p(exp(-|z|))
    ls += fminf(z, 0.f) - log1pf(expf(-fabsf(z)));
    sc += 1.f / (1.f + expf(-zc));
    sr += 1.f / (1.f + expf(-zr));
  }
  const float ls_sum = block_sum_f(ls, tid, sh_f);
  const float sc_sum = block_sum_f(sc, tid, sh_f);
  const float sr_sum = block_sum_f(sr, tid, sh_f);

  if (tid == 0) {
    int mcount = end_ind - div_ind;
    if (mcount < 0) mcount = 0;
    const float cnt = fmaxf((float)mcount, 1.f);
    partials[b]          = -(ls_sum / cnt);                       // loss contribution
    const float dsc      = (sc_sum - sr_sum) / cnt;
    partials[BPAIRS + b] = (dsc > 0.5f) ? 1.f : 0.f;              // acc contribution
    out[2 + b]           = rc[(c_ind - 1 + SEQ) % SEQ];           // chosen_mean_score
    out[2 + BPAIRS + b]  = rr[(r_ind - 1 + SEQ) % SEQ];           // rejected_mean_score
  }
}

// ---------------------------------------------------------------------------
// Phase 3: fold 32 per-pair partials into loss / acc scalars.
// ---------------------------------------------------------------------------
__global__ void finalize(const float* __restrict__ partials, float* __restrict__ out) {
  if (threadIdx.x == 0) {
    float l = 0.f, a = 0.f;
    for (int i = 0; i < BPAIRS; ++i) {
      l += partials[i];
      a += partials[BPAIRS + i];
    }
    out[0] = l;
    out[1] = a;
  }
}

extern "C" void kernel_launch(void* const* d_in, const int* in_sizes, int n_in,
                              void* d_out, int out_size, void* d_ws, size_t ws_size,
                              hipStream_t stream) {
  const int*   ids    = (const int*)d_in[0];
  const float* hidden = (const float*)d_in[1];
  const float* w      = (const float*)d_in[2];
  float* out      = (float*)d_out;
  float* rewards  = (float*)d_ws;                 // 2*B*S floats = 512 KB
  float* partials = rewards + 2 * BPAIRS * SEQ;   // 64 floats

  // 131072 rows total; 128 rows per block (8 waves x 16-row WMMA tiles)
  const int nblocks = (2 * BPAIRS * SEQ) / (16 * 8);
  gemv_rewards_wmma<<<nblocks, 256, 0, stream>>>(hidden, w, rewards);
  pair_reduce<<<BPAIRS, 256, 0, stream>>>(ids, rewards, out, partials);
  finalize<<<1, 32, 0, stream>>>(partials, out);
}